// TtMixtralAttention_42021960024274
// MI455X (gfx1250) — compile-verified
//
#include <hip/hip_runtime.h>
#include <math.h>

// ---- problem constants (match reference) ----
#define HIDDEN   4096
#define QKV_DIM  6144
#define NHEADS   32
#define NKV      8
#define GQA      4              // NHEADS / NKV
#define HD       128
#define BATCH    32
#define CLEN     4096
#define QD       (NHEADS * HD)          // 4096
#define KDOFF    (QD + NKV * HD)        // 5120 (v offset inside xqkv row)

// attention decomposition: NSPLIT blocks x WAVES waves per (b, kv-head)
#define NSPLIT   8
#define WAVES    4
#define NPART    NSPLIT                  // one partial per block (LDS-combined)
#define ROWS_PER_WAVE (CLEN / (NSPLIT * WAVES))   // 128

typedef __attribute__((ext_vector_type(2))) float v2f;
typedef __attribute__((ext_vector_type(4))) float v4f;
typedef __attribute__((ext_vector_type(8))) float v8f;

__device__ __forceinline__ v8f wmma_f32_16x16x4(v2f a, v2f b, v8f c) {
  // 8 args: (neg_a, A, neg_b, B, c_mod, C, reuse_a, reuse_b)
  return __builtin_amdgcn_wmma_f32_16x16x4_f32(false, a, false, b, (short)0, c,
                                               false, false);
}

// -----------------------------------------------------------------------------
// C (32 x N) = A (32 x 4096) @ W (4096 x N), fp32 WMMA 16x16x4.
// One wave per 16-wide N tile; two accumulators cover M=0..15 and M=16..31.
// Regular (RT) loads on purpose: wqkv+wo (167 MB) can stay L2-resident across
// graph replays; the KV stream below uses NT so it does not evict them.
// -----------------------------------------------------------------------------
__global__ __launch_bounds__(32) void gemm32_wmma_kernel(
    const float* __restrict__ A, const float* __restrict__ W,
    float* __restrict__ C, int N) {
  const int nt   = blockIdx.x;
  const int lane = threadIdx.x & 31;
  const int lm   = lane & 15;
  const int kofs = (lane >> 4) << 1;     // 0 for lanes 0-15, 2 for lanes 16-31

  const float* ar0 = A + (size_t)lm * HIDDEN;
  const float* ar1 = A + (size_t)(16 + lm) * HIDDEN;
  const float* wc  = W + (size_t)nt * 16 + lm;

  v8f acc0 = {};
  v8f acc1 = {};
#pragma unroll 4
  for (int k = 0; k < HIDDEN; k += 4) {
    const int kk = k + kofs;
    v2f a0 = {ar0[kk], ar0[kk + 1]};
    v2f a1 = {ar1[kk], ar1[kk + 1]};
    v2f b  = {wc[(size_t)kk * N], wc[(size_t)(kk + 1) * N]};
    acc0 = wmma_f32_16x16x4(a0, b, acc0);
    acc1 = wmma_f32_16x16x4(a1, b, acc1);
  }

  // D layout: VGPR j -> row j (lanes 0-15) / row j+8 (lanes 16-31), col = lane&15
  const int rofs = (lane >> 4) * 8;
  const int col  = nt * 16 + lm;
#pragma unroll
  for (int j = 0; j < 8; ++j) {
    C[(size_t)(rofs + j) * N + col]      = acc0[j];
    C[(size_t)(16 + rofs + j) * N + col] = acc1[j];
  }
}

// -----------------------------------------------------------------------------
// RoPE as GEMM: rows r in [0, B*40): b = r/40, h = r%40.
// h < 32 -> q head h; h >= 32 -> k head (h-32). Output qk: (B, 40, 128).
// -----------------------------------------------------------------------------
__global__ __launch_bounds__(32) void rope_wmma_kernel(
    const float* __restrict__ xqkv, const float* __restrict__ rot,
    float* __restrict__ qk) {
  const int mt   = blockIdx.x;  // 0..79  (1280 rows / 16)
  const int nt   = blockIdx.y;  // 0..7   (128 cols / 16)
  const int lane = threadIdx.x & 31;
  const int lm   = lane & 15;
  const int kofs = (lane >> 4) << 1;

  const int r  = mt * 16 + lm;
  const int bb = r / 40;
  const int h  = r % 40;
  const float* arow =
      xqkv + (size_t)bb * QKV_DIM + (h < 32 ? h * HD : QD + (h - 32) * HD);
  const float* rc = rot + nt * 16 + lm;

  v8f acc = {};
#pragma unroll
  for (int k = 0; k < HD; k += 4) {
    const int kk = k + kofs;
    v2f a = {arow[kk], arow[kk + 1]};
    v2f b = {rc[(size_t)kk * HD], rc[(size_t)(kk + 1) * HD]};
    acc = wmma_f32_16x16x4(a, b, acc);
  }

  const int rofs = (lane >> 4) * 8;
  const int col  = nt * 16 + lm;
#pragma unroll
  for (int j = 0; j < 8; ++j) {
    const int row = mt * 16 + rofs + j;  // == b*40 + h for the output element
    qk[(size_t)row * HD + col] = acc[j];
  }
}

// -----------------------------------------------------------------------------
// Streaming GQA flash attention. One wave per 128-row chunk of L (8192 waves
// device-wide). KV rows loaded with NON-TEMPORAL hint (1 GB stream; keep L2
// for the weights). Row current_pos substituted with fresh RoPE'd k / raw v
// via branch-free loop splitting. 4 waves per block are combined in LDS so
// each block emits a single softmax partial.
// -----------------------------------------------------------------------------
__global__ __launch_bounds__(128) void attn_partial_kernel(
    const float* __restrict__ qk, const float* __restrict__ xqkv,
    const float* __restrict__ cache_k, const float* __restrict__ cache_v,
    const float* __restrict__ mask, const int* __restrict__ cur_pos,
    float* __restrict__ part_m, float* __restrict__ part_s,
    float* __restrict__ part_out) {
  const int bid  = blockIdx.x;               // B * NKV * NSPLIT
  const int sp   = bid % NSPLIT;
  const int kv   = (bid / NSPLIT) % NKV;
  const int b    = bid / (NSPLIT * NKV);
  const int wave = threadIdx.x >> 5;
  const int lane = threadIdx.x & 31;
  const int l0   = (sp * WAVES + wave) * ROWS_PER_WAVE;
  const int lend = l0 + ROWS_PER_WAVE;
  const int d4   = lane * 4;
  const int cur  = cur_pos[0];

  v4f qv[GQA];
#pragma unroll
  for (int g = 0; g < GQA; ++g)
    qv[g] = *(const v4f*)(qk + (size_t)(b * 40 + kv * GQA + g) * HD + d4);

  const v4f knew = *(const v4f*)(qk + (size_t)(b * 40 + 32 + kv) * HD + d4);
  const v4f vnew =
      *(const v4f*)(xqkv + (size_t)b * QKV_DIM + KDOFF + kv * HD + d4);

  const float* kbase = cache_k + (size_t)(b * NKV + kv) * CLEN * HD + d4;
  const float* vbase = cache_v + (size_t)(b * NKV + kv) * CLEN * HD + d4;
  const float* mb    = mask + (size_t)(kv * GQA) * BATCH * CLEN + (size_t)b * CLEN;

  float m[GQA], s[GQA];
  v4f   acc[GQA];
#pragma unroll
  for (int g = 0; g < GQA; ++g) {
    m[g]   = -1e30f;
    s[g]   = 0.f;
    acc[g] = (v4f)0.f;
  }

  const float scale = 0.088388347648318447f;  // 1/sqrt(128)

  auto process = [&](int l, v4f kr, v4f vr) {
    float sc[GQA];
#pragma unroll
    for (int g = 0; g < GQA; ++g) {
      v4f t = qv[g] * kr;
      float pd = (t.x + t.y) + (t.z + t.w);
#pragma unroll
      for (int off = 16; off > 0; off >>= 1) pd += __shfl_xor(pd, off, 32);
      sc[g] = pd * scale + mb[(size_t)g * BATCH * CLEN + l];
    }
#pragma unroll
    for (int g = 0; g < GQA; ++g) {
      const float mn = fmaxf(m[g], sc[g]);
      const float co = __expf(m[g] - mn);
      const float pe = __expf(sc[g] - mn);
      s[g]   = s[g] * co + pe;
      acc[g] = acc[g] * co + vr * pe;
      m[g]   = mn;
    }
  };

  auto stream_row = [&](int l) {
    v4f kr = __builtin_nontemporal_load((const v4f*)(kbase + (size_t)l * HD));
    v4f vr = __builtin_nontemporal_load((const v4f*)(vbase + (size_t)l * HD));
    process(l, kr, vr);
  };

  const bool has_cur = (cur >= l0) && (cur < lend);
  const int  mid     = has_cur ? cur : lend;

#pragma unroll 2
  for (int l = l0; l < mid; ++l) stream_row(l);
  if (has_cur) {
    process(cur, knew, vnew);
#pragma unroll 2
    for (int l = cur + 1; l < lend; ++l) stream_row(l);
  }

  // ---- combine the 4 waves of this block in LDS -> one partial per block ----
  __shared__ float lds_m[WAVES][GQA];
  __shared__ float lds_s[WAVES][GQA];
  __shared__ v4f   lds_acc[WAVES][GQA][HD / 4];  // 8 KB

#pragma unroll
  for (int g = 0; g < GQA; ++g) {
    if (lane == 0) {
      lds_m[wave][g] = m[g];
      lds_s[wave][g] = s[g];
    }
    lds_acc[wave][g][lane] = acc[g];
  }
  __syncthreads();

  if (wave == 0) {
    const size_t pidx = ((size_t)(b * NKV + kv) * NPART + sp) * GQA;
#pragma unroll
    for (int g = 0; g < GQA; ++g) {
      float M = lds_m[0][g];
#pragma unroll
      for (int w = 1; w < WAVES; ++w) M = fmaxf(M, lds_m[w][g]);
      float S = 0.f;
      v4f   O = (v4f)0.f;
#pragma unroll
      for (int w = 0; w < WAVES; ++w) {
        const float f = __expf(lds_m[w][g] - M);
        S += f * lds_s[w][g];
        O += lds_acc[w][g][lane] * f;
      }
      if (lane == 0) {
        part_m[pidx + g] = M;
        part_s[pidx + g] = S;
      }
      *(v4f*)(part_out + (pidx + g) * HD + d4) = O;
    }
  }
}

// -----------------------------------------------------------------------------
// Combine the NPART softmax partials per (b, kv, g); write concat-heads layout.
// -----------------------------------------------------------------------------
__global__ __launch_bounds__(32) void attn_combine_kernel(
    const float* __restrict__ part_m, const float* __restrict__ part_s,
    const float* __restrict__ part_out, float* __restrict__ attn_cat) {
  const int bid  = blockIdx.x;  // B * NKV * GQA
  const int g    = bid % GQA;
  const int kv   = (bid / GQA) % NKV;
  const int b    = bid / (GQA * NKV);
  const int lane = threadIdx.x & 31;
  const int d4   = lane * 4;

  const size_t base = (size_t)(b * NKV + kv) * NPART * GQA + g;
  float M = -1e30f;
#pragma unroll
  for (int p = 0; p < NPART; ++p) M = fmaxf(M, part_m[base + (size_t)p * GQA]);

  float S = 0.f;
  v4f   o = (v4f)0.f;
#pragma unroll
  for (int p = 0; p < NPART; ++p) {
    const size_t idx = base + (size_t)p * GQA;
    const float  w   = __expf(part_m[idx] - M);
    S += w * part_s[idx];
    o += *(const v4f*)(part_out + idx * HD + d4) * w;
  }
  const float inv = 1.f / S;
  *(v4f*)(attn_cat + (size_t)b * HIDDEN + (size_t)(kv * GQA + g) * HD + d4) =
      o * inv;
}

// -----------------------------------------------------------------------------
extern "C" void kernel_launch(void* const* d_in, const int* in_sizes, int n_in,
                              void* d_out, int out_size, void* d_ws,
                              size_t ws_size, hipStream_t stream) {
  const float* x    = (const float*)d_in[0];  // (1,1,32,4096)
  const float* wqkv = (const float*)d_in[1];  // (4096,6144)
  const float* wo   = (const float*)d_in[2];  // (4096,4096)
  const float* rot  = (const float*)d_in[3];  // (128,128)
  const float* ck   = (const float*)d_in[4];  // (32,8,4096,128)
  const float* cv   = (const float*)d_in[5];  // (32,8,4096,128)
  const float* mask = (const float*)d_in[6];  // (1,32,32,4096)
  const int*   cur  = (const int*)d_in[7];    // scalar
  float*       out  = (float*)d_out;          // (32,1,1,4096) == (32,4096)

  // workspace layout (floats); total ~6.2 MB
  float* ws   = (float*)d_ws;
  float* xqkv = ws;                                   // 32*6144    = 196608
  float* qk   = xqkv + (size_t)BATCH * QKV_DIM;       // 32*40*128  = 163840
  float* pm   = qk + (size_t)BATCH * 40 * HD;         // 32*8*8*4   = 8192
  float* ps   = pm + (size_t)BATCH * NKV * NPART * GQA;
  float* po   = ps + (size_t)BATCH * NKV * NPART * GQA;  // *128    = 1048576
  float* cat  = po + (size_t)BATCH * NKV * NPART * GQA * HD;  // 32*4096

  // 1) fused QKV projection (WMMA f32)
  gemm32_wmma_kernel<<<QKV_DIM / 16, 32, 0, stream>>>(x, wqkv, xqkv, QKV_DIM);

  // 2) RoPE for q and k heads (WMMA f32)
  dim3 rg((BATCH * (NHEADS + NKV)) / 16, HD / 16);
  rope_wmma_kernel<<<rg, 32, 0, stream>>>(xqkv, rot, qk);

  // 3) streaming flash attention over the KV cache (bandwidth-critical, NT)
  attn_partial_kernel<<<BATCH * NKV * NSPLIT, WAVES * 32, 0, stream>>>(
      qk, xqkv, ck, cv, mask, cur, pm, ps, po);

  // 4) combine softmax partials -> concat heads
  attn_combine_kernel<<<BATCH * NKV * GQA, 32, 0, stream>>>(pm, ps, po, cat);

  // 5) output projection (WMMA f32), final permute is a no-op on (32,4096)
  gemm32_wmma_kernel<<<HIDDEN / 16, 32, 0, stream>>>(cat, wo, out, HIDDEN);
}